// MeshConv_49263274885412
// MI455X (gfx1250) — compile-verified
//
#include <hip/hip_runtime.h>

#define NVV   40962
#define NFF   81920
#define BB    16
#define CIN_  32
#define COUT_ 32
#define NTILES ((NVV + 15) / 16)   // 2561

typedef float v2f __attribute__((ext_vector_type(2)));
typedef float v8f __attribute__((ext_vector_type(8)));

__global__ __launch_bounds__(64)
void MeshConv_fused_wmma(const float* __restrict__ x,
                         const int*   __restrict__ G_cols,  const float* __restrict__ G_vals,
                         const int*   __restrict__ L_cols,  const float* __restrict__ L_vals,
                         const int*   __restrict__ F2V_cols,const float* __restrict__ F2V_vals,
                         const float* __restrict__ EW,      const float* __restrict__ NS,
                         const float* __restrict__ coeffs,  const float* __restrict__ bias,
                         float* __restrict__ out)
{
    __shared__ float feat_lds[16 * 128];      //  8 KB : feat[vtx][c*4+k]
    __shared__ float coef_lds[COUT_ * 128];   // 16 KB : coeffs[o][c*4+k]

    const int t     = threadIdx.x;
    const int tile  = blockIdx.x;
    const int b     = blockIdx.y;
    const int vbase = tile * 16;

    // coeffs are already [COUT][CIN][4] = [o][kappa] flat; coalesced preload
    for (int i = t; i < COUT_ * 128; i += 64) coef_lds[i] = coeffs[i];

    // ---------------- Phase 1: build feat[16][128] in LDS ----------------
    const int vloc = t & 15;         // vertex within tile
    const int cgrp = t >> 4;         // 0..3 -> channels cgrp*8 .. cgrp*8+7
    int n = vbase + vloc;
    if (n >= NVV) n = NVV - 1;       // clamp; masked at store

    const float* xb = x + (size_t)b * CIN_ * NVV;

    // Laplacian indices: L has exactly 7 nnz per row, rows contiguous
    int   lcol[7]; float lval[7];
    #pragma unroll
    for (int i = 0; i < 7; ++i) { lcol[i] = L_cols[n*7 + i]; lval[i] = L_vals[n*7 + i]; }

    float ewacc[8], nsacc[8];
    #pragma unroll
    for (int ci = 0; ci < 8; ++ci) {
        const int c = cgrp*8 + ci;
        const float* xc = xb + (size_t)c * NVV;
        float lap = 0.f;
        #pragma unroll
        for (int i = 0; i < 7; ++i) lap += lval[i] * xc[lcol[i]];
        feat_lds[vloc*128 + c*4 + 0] = xc[n];   // identity
        feat_lds[vloc*128 + c*4 + 1] = lap;     // laplacian
        ewacc[ci] = 0.f; nsacc[ci] = 0.f;
    }

    // F2V: exactly 6 faces per vertex; G: row r = comp*NF + f, 3 nnz at 3r..3r+2
    for (int j = 0; j < 6; ++j) {
        const int   f = F2V_cols[n*6 + j];
        const float w = F2V_vals[n*6 + j];
        int gc[9]; float gv[9];
        #pragma unroll
        for (int k = 0; k < 3; ++k) {
            const int base = 3 * (k*NFF + f);
            #pragma unroll
            for (int i = 0; i < 3; ++i) { gc[k*3+i] = G_cols[base+i]; gv[k*3+i] = G_vals[base+i]; }
        }
        const float e0 = EW[f*3+0], e1 = EW[f*3+1], e2 = EW[f*3+2];
        const float s0 = NS[f*3+0], s1 = NS[f*3+1], s2 = NS[f*3+2];
        #pragma unroll
        for (int ci = 0; ci < 8; ++ci) {
            const int c = cgrp*8 + ci;
            const float* xc = xb + (size_t)c * NVV;
            const float g0 = gv[0]*xc[gc[0]] + gv[1]*xc[gc[1]] + gv[2]*xc[gc[2]];
            const float g1 = gv[3]*xc[gc[3]] + gv[4]*xc[gc[4]] + gv[5]*xc[gc[5]];
            const float g2 = gv[6]*xc[gc[6]] + gv[7]*xc[gc[7]] + gv[8]*xc[gc[8]];
            ewacc[ci] += w * (g0*e0 + g1*e1 + g2*e2);
            nsacc[ci] += w * (g0*s0 + g1*s1 + g2*s2);
        }
    }
    #pragma unroll
    for (int ci = 0; ci < 8; ++ci) {
        const int c = cgrp*8 + ci;
        feat_lds[vloc*128 + c*4 + 2] = ewacc[ci];
        feat_lds[vloc*128 + c*4 + 3] = nsacc[ci];
    }

    __syncthreads();

    // ------- Phase 2: D[o=16][v=16] += A(coeffs 16x4) x B(feat 4x16) -------
    // wave32: 2 waves per block, one 16-output-channel tile each
    const int wave  = t >> 5;
    const int obase = wave * 16;
    const int lane  = t & 31;
    const int half  = lane >> 4;   // lanes 0-15 hold K={0,1}, lanes 16-31 K={2,3}
    const int mrow  = lane & 15;

    v8f acc = {0.f,0.f,0.f,0.f,0.f,0.f,0.f,0.f};
    #pragma unroll
    for (int s = 0; s < 32; ++s) {
        const int kk = 4*s + 2*half;
        v2f Af = *(const v2f*)&coef_lds[(obase + mrow)*128 + kk];  // A[M=o][K]
        v2f Bf = *(const v2f*)&feat_lds[mrow*128 + kk];            // B[K][N=v]
        acc = __builtin_amdgcn_wmma_f32_16x16x4_f32(false, Af, false, Bf,
                                                    (short)0, acc, false, false);
    }

    // D layout: VGPR i -> M = i + 8*half, N = mrow  => coalesced along NV
    const int vcol = vbase + mrow;
    if (vcol < NVV) {
        #pragma unroll
        for (int i = 0; i < 8; ++i) {
            const int o = obase + i + 8*half;
            out[((size_t)b * COUT_ + o) * (size_t)NVV + vcol] = acc[i] + bias[o];
        }
    }
}

extern "C" void kernel_launch(void* const* d_in, const int* in_sizes, int n_in,
                              void* d_out, int out_size, void* d_ws, size_t ws_size,
                              hipStream_t stream) {
    (void)in_sizes; (void)n_in; (void)out_size; (void)d_ws; (void)ws_size;
    const float* x        = (const float*)d_in[0];
    // d_in[1] = G_rows (structure known: repeat(arange(3*NF), 3)) -- unused
    const int*   G_cols   = (const int*)  d_in[2];
    const float* G_vals   = (const float*)d_in[3];
    // d_in[4] = L_rows (repeat(arange(NV), 7)) -- unused
    const int*   L_cols   = (const int*)  d_in[5];
    const float* L_vals   = (const float*)d_in[6];
    // d_in[7] = F2V_rows (repeat(arange(NV), 6)) -- unused
    const int*   F2V_cols = (const int*)  d_in[8];
    const float* F2V_vals = (const float*)d_in[9];
    const float* EW       = (const float*)d_in[10];
    const float* NS       = (const float*)d_in[11];
    const float* coeffs   = (const float*)d_in[12];
    const float* bias     = (const float*)d_in[13];
    float*       out      = (float*)d_out;

    dim3 grid(NTILES, BB);
    MeshConv_fused_wmma<<<grid, 64, 0, stream>>>(x, G_cols, G_vals, L_cols, L_vals,
                                                 F2V_cols, F2V_vals, EW, NS,
                                                 coeffs, bias, out);
}